// GAT_38130719654057
// MI455X (gfx1250) — compile-verified
//
#include <hip/hip_runtime.h>

#define NEG_SLOPE 0.2f
#define BN_EPS    1e-5f

typedef __attribute__((ext_vector_type(2))) float v2f;
typedef __attribute__((ext_vector_type(8))) float v8f;

// ---------------------------------------------------------------------------
// WMMA f32 GEMM: C[M,NC] = A[M,K] * B[K,NC].  One 16x64 strip per wave
// (4 accumulators share one A fragment), V_WMMA_F32_16X16X4_F32, K step 4.
// M, K multiples of 16/4; NC multiple of 64.
// ---------------------------------------------------------------------------
__global__ __launch_bounds__(256) void wmma_gemm_f32(
    const float* __restrict__ A, const float* __restrict__ B,
    float* __restrict__ C, int M, int K, int NC)
{
    const int lane = threadIdx.x & 31;
    const int wave = threadIdx.x >> 5;
    const int ng   = NC >> 6;                      // 64-col groups
    const int mtiles = M >> 4;
    const int grp = blockIdx.x * 8 + wave;
    if (grp >= mtiles * ng) return;                // wave-uniform: EXEC stays all-1
    const int mt = grp / ng, nt4 = grp % ng;
    const int half = lane >> 4, l15 = lane & 15;

    const float* arow  = A + (size_t)(mt * 16 + l15) * K;   // A frag: K0/K1 low half, K2/K3 high half
    const float* bbase = B + nt4 * 64 + l15;

    v8f acc0 = {}, acc1 = {}, acc2 = {}, acc3 = {};
    #pragma unroll 2
    for (int kb = 0; kb < K; kb += 4) {
        const int k0 = kb + 2 * half;                        // even -> 8B aligned
        v2f a = *(const v2f*)(arow + k0);                    // one b64 load feeds 4 WMMAs
        const float* b0 = bbase + (size_t)k0 * NC;
        const float* b1 = bbase + (size_t)(k0 + 1) * NC;
        v2f b;
        b.x = b0[0];  b.y = b1[0];
        acc0 = __builtin_amdgcn_wmma_f32_16x16x4_f32(false, a, false, b, (short)0, acc0, false, false);
        b.x = b0[16]; b.y = b1[16];
        acc1 = __builtin_amdgcn_wmma_f32_16x16x4_f32(false, a, false, b, (short)0, acc1, false, false);
        b.x = b0[32]; b.y = b1[32];
        acc2 = __builtin_amdgcn_wmma_f32_16x16x4_f32(false, a, false, b, (short)0, acc2, false, false);
        b.x = b0[48]; b.y = b1[48];
        acc3 = __builtin_amdgcn_wmma_f32_16x16x4_f32(false, a, false, b, (short)0, acc3, false, false);
    }
    float* crow = C + (size_t)(mt * 16 + half * 8) * NC + nt4 * 64 + l15;
    #pragma unroll
    for (int v = 0; v < 8; ++v) {                 // C layout: M = v + 8*half, N = l15 (+16k)
        crow[(size_t)v * NC +  0] = acc0[v];
        crow[(size_t)v * NC + 16] = acc1[v];
        crow[(size_t)v * NC + 32] = acc2[v];
        crow[(size_t)v * NC + 48] = acc3[v];
    }
}

// ---------------------------------------------------------------------------
// CSR build (dst-grouped), reused by all 3 layers.
// ---------------------------------------------------------------------------
__global__ void zero_u32(unsigned int* __restrict__ p, int n)
{
    int i = blockIdx.x * blockDim.x + threadIdx.x;
    if (i < n) p[i] = 0u;
}

__global__ void degree_count(const int* __restrict__ edge_dst, int* __restrict__ deg,
                             int nNodes, int E)
{
    int i = blockIdx.x * blockDim.x + threadIdx.x;
    if (i >= E + nNodes) return;
    int d = (i < E) ? edge_dst[i] : (i - E);       // self-loop appended
    atomicAdd(&deg[d], 1);
}

__global__ __launch_bounds__(1024) void scan_rowptr(const int* __restrict__ deg,
                                                    int* __restrict__ rowptr, int n)
{
    __shared__ int buf[1024];
    __shared__ int carry;
    const int tid = threadIdx.x;
    if (tid == 0) carry = 0;
    __syncthreads();
    for (int base = 0; base < n; base += 1024) {
        int i = base + tid;
        int v = (i < n) ? deg[i] : 0;
        buf[tid] = v;
        __syncthreads();
        for (int off = 1; off < 1024; off <<= 1) {
            int t = (tid >= off) ? buf[tid - off] : 0;
            __syncthreads();
            buf[tid] += t;
            __syncthreads();
        }
        if (i < n) rowptr[i] = carry + buf[tid] - v;   // exclusive
        __syncthreads();
        if (tid == 0) carry += buf[1023];
        __syncthreads();
    }
    if (tid == 0) rowptr[n] = carry;
}

__global__ void copy_i32(const int* __restrict__ a, int* __restrict__ b, int n)
{
    int i = blockIdx.x * blockDim.x + threadIdx.x;
    if (i < n) b[i] = a[i];
}

__global__ void csr_fill(const int* __restrict__ edge_dst, int* __restrict__ cursor,
                         int* __restrict__ eidx, int nNodes, int E)
{
    int i = blockIdx.x * blockDim.x + threadIdx.x;
    if (i >= E + nNodes) return;
    int d = (i < E) ? edge_dst[i] : (i - E);
    int pos = atomicAdd(&cursor[d], 1);
    eidx[pos] = i;                                 // encodes src: <E => edge_src[i], else i-E
}

// ---------------------------------------------------------------------------
// Attention scores: s_src[n,h] = <h[n,h,:], a_src[h,:]>, likewise s_dst.
// One wave per (node, head); C = 128 = 32 lanes x float4.
// ---------------------------------------------------------------------------
template <int H>
__global__ void gat_scores(const float* __restrict__ hW,
                           const float* __restrict__ asrc, const float* __restrict__ adst,
                           float* __restrict__ s_src, float* __restrict__ s_dst, int nNodes)
{
    const int n = blockIdx.x;
    const int lane = threadIdx.x & 31;
    const int h = threadIdx.x >> 5;
    const float4* row = (const float4*)(hW + (size_t)n * (H * 128) + h * 128);
    const float4* as  = (const float4*)(asrc + h * 128);
    const float4* ad  = (const float4*)(adst + h * 128);
    float4 v = row[lane], a = as[lane], b = ad[lane];
    float ss = v.x * a.x + v.y * a.y + v.z * a.z + v.w * a.w;
    float sd = v.x * b.x + v.y * b.y + v.z * b.z + v.w * b.w;
    #pragma unroll
    for (int off = 16; off; off >>= 1) {
        ss += __shfl_xor(ss, off, 32);
        sd += __shfl_xor(sd, off, 32);
    }
    if (lane == 0) { s_src[n * H + h] = ss; s_dst[n * H + h] = sd; }
}

// ---------------------------------------------------------------------------
// Per-dst softmax + aggregation.  One wave per destination node; float4
// register accumulators (H per lane); b128 gathers; no float atomics.
// ---------------------------------------------------------------------------
template <int H>
__global__ __launch_bounds__(256) void gat_aggregate(
    const float* __restrict__ hW, const float* __restrict__ ssrc,
    const float* __restrict__ sdstArr, const int* __restrict__ rowptr,
    const int* __restrict__ eidx, const int* __restrict__ edge_src,
    float* __restrict__ outb, int nNodes, int E)
{
    constexpr int OC = H * 128;
    const int d = blockIdx.x * 8 + (threadIdx.x >> 5);
    if (d >= nNodes) return;
    const int lane = threadIdx.x & 31;
    const int start = rowptr[d], end = rowptr[d + 1];

    float sdst[H], mx[H], den[H];
    #pragma unroll
    for (int h = 0; h < H; ++h) { sdst[h] = sdstArr[d * H + h]; mx[h] = -3.0e38f; den[h] = 0.f; }

    // pass 1: segment max of leaky_relu(s_src[src]+s_dst[d])
    for (int p = start + lane; p < end; p += 32) {
        int e = eidx[p];
        int src = (e < E) ? edge_src[e] : (e - E);
        #pragma unroll
        for (int h = 0; h < H; ++h) {
            float t = ssrc[src * H + h] + sdst[h];
            t = (t > 0.f) ? t : NEG_SLOPE * t;
            mx[h] = fmaxf(mx[h], t);
        }
    }
    #pragma unroll
    for (int off = 16; off; off >>= 1)
        #pragma unroll
        for (int h = 0; h < H; ++h) mx[h] = fmaxf(mx[h], __shfl_xor(mx[h], off, 32));

    // pass 2: denominator
    for (int p = start + lane; p < end; p += 32) {
        int e = eidx[p];
        int src = (e < E) ? edge_src[e] : (e - E);
        #pragma unroll
        for (int h = 0; h < H; ++h) {
            float t = ssrc[src * H + h] + sdst[h];
            t = (t > 0.f) ? t : NEG_SLOPE * t;
            den[h] += __expf(t - mx[h]);
        }
    }
    #pragma unroll
    for (int off = 16; off; off >>= 1)
        #pragma unroll
        for (int h = 0; h < H; ++h) den[h] += __shfl_xor(den[h], off, 32);

    float inv[H];
    #pragma unroll
    for (int h = 0; h < H; ++h) inv[h] = 1.f / den[h];     // self-loop => den > 0

    // per-lane selects for the lane<H alpha computation (no dynamic reg indexing)
    float sd_l = sdst[0], mx_l = mx[0], inv_l = inv[0];
    #pragma unroll
    for (int h = 1; h < H; ++h)
        if (lane == h) { sd_l = sdst[h]; mx_l = mx[h]; inv_l = inv[h]; }

    // pass 3: out[d] = sum_e alpha_e * h[src_e]   (chunk h = cols h*128..h*128+127)
    float4 acc[H];
    #pragma unroll
    for (int h = 0; h < H; ++h) acc[h] = make_float4(0.f, 0.f, 0.f, 0.f);

    int src = 0;
    if (start < end) { int e = eidx[start]; src = (e < E) ? edge_src[e] : (e - E); }
    for (int p = start; p < end; ++p) {
        int srcn = src;
        if (p + 1 < end) {
            int en = eidx[p + 1];
            srcn = (en < E) ? edge_src[en] : (en - E);
            __builtin_prefetch(hW + (size_t)srcn * OC + lane * (OC / 8), 0, 1);  // cover full row
        }
        float alpha = 0.f;
        if (lane < H) {
            float t = ssrc[src * H + lane] + sd_l;
            t = (t > 0.f) ? t : NEG_SLOPE * t;
            alpha = __expf(t - mx_l) * inv_l;
        }
        float al[H];
        #pragma unroll
        for (int h = 0; h < H; ++h) al[h] = __shfl(alpha, h, 32);
        const float4* hs = (const float4*)(hW + (size_t)src * OC);
        #pragma unroll
        for (int h = 0; h < H; ++h) {
            float4 v = hs[h * 32 + lane];                   // b128 gather
            acc[h].x = fmaf(al[h], v.x, acc[h].x);
            acc[h].y = fmaf(al[h], v.y, acc[h].y);
            acc[h].z = fmaf(al[h], v.z, acc[h].z);
            acc[h].w = fmaf(al[h], v.w, acc[h].w);
        }
        src = srcn;
    }
    float4* orow = (float4*)(outb + (size_t)d * OC);
    #pragma unroll
    for (int h = 0; h < H; ++h) orow[h * 32 + lane] = acc[h];
}

// ---------------------------------------------------------------------------
// BatchNorm (training-mode batch stats) + ELU.  Bias cancels inside BN.
// ---------------------------------------------------------------------------
__global__ void bn_stats(const float* __restrict__ x, float* __restrict__ sum,
                         float* __restrict__ sq, int nRows, int OC)
{
    const int col = blockIdx.y * 128 + threadIdx.x;
    float s = 0.f, ss = 0.f;
    for (int r = blockIdx.x; r < nRows; r += gridDim.x) {
        float v = x[(size_t)r * OC + col];
        s += v; ss += v * v;
    }
    atomicAdd(&sum[col], s);
    atomicAdd(&sq[col], ss);
}

__global__ void bn_apply_elu(float* __restrict__ x, const float* __restrict__ sum,
                             const float* __restrict__ sq, const float* __restrict__ gamma,
                             const float* __restrict__ beta, int nRows, int OC)
{
    size_t i = (size_t)blockIdx.x * blockDim.x + threadIdx.x;
    if (i >= (size_t)nRows * OC) return;
    int col = (int)(i % OC);
    float invN = 1.f / (float)nRows;
    float mu  = sum[col] * invN;
    float var = sq[col] * invN - mu * mu;
    float y = gamma[col] * (x[i] - mu) * rsqrtf(var + BN_EPS) + beta[col];
    x[i] = (y > 0.f) ? y : (__expf(y) - 1.f);
}

// ---------------------------------------------------------------------------
// Global mean pool over batch_idx.
// ---------------------------------------------------------------------------
__global__ void pool_accum(const float* __restrict__ act, const int* __restrict__ batch,
                           float* __restrict__ pool, float* __restrict__ cnt, int nNodes)
{
    const int n = blockIdx.x, t = threadIdx.x;
    const int b = batch[n];
    atomicAdd(&pool[b * 128 + t], act[(size_t)n * 128 + t]);
    if (t == 0) atomicAdd(&cnt[b], 1.0f);
}

__global__ void pool_final(const float* __restrict__ pool, const float* __restrict__ cnt,
                           float* __restrict__ out, int G)
{
    int i = blockIdx.x * blockDim.x + threadIdx.x;
    if (i >= G * 128) return;
    out[i] = pool[i] / fmaxf(cnt[i / 128], 1.0f);
}

// ---------------------------------------------------------------------------
extern "C" void kernel_launch(void* const* d_in, const int* in_sizes, int n_in,
                              void* d_out, int out_size, void* d_ws, size_t ws_size,
                              hipStream_t stream)
{
    const float* x     = (const float*)d_in[0];
    const int*   ei    = (const int*)d_in[1];
    const int*   batch = (const int*)d_in[2];
    const float* W[3]    = { (const float*)d_in[3],  (const float*)d_in[9],  (const float*)d_in[15] };
    const float* ASRC[3] = { (const float*)d_in[4],  (const float*)d_in[10], (const float*)d_in[16] };
    const float* ADST[3] = { (const float*)d_in[5],  (const float*)d_in[11], (const float*)d_in[17] };
    const float* GAM[3]  = { (const float*)d_in[7],  (const float*)d_in[13], (const float*)d_in[19] };
    const float* BET[3]  = { (const float*)d_in[8],  (const float*)d_in[14], (const float*)d_in[20] };

    const int N = in_sizes[0] / 128;
    const int E = in_sizes[1] / 2;
    const int G = out_size / 128;
    const int Etot = E + N;
    const int* esrc = ei;
    const int* edst = ei + E;

    char* wp = (char*)d_ws;
    auto carve = [&](size_t bytes) -> void* {
        void* p = (void*)wp;
        wp += (bytes + 255) & ~(size_t)255;
        return p;
    };
    float* bufA   = (float*)carve((size_t)N * 512 * 4);   // GEMM out / gather source
    float* bufB   = (float*)carve((size_t)N * 512 * 4);   // aggregation out (BN in place)
    float* ssrc   = (float*)carve((size_t)N * 4 * 4);
    float* sdst   = (float*)carve((size_t)N * 4 * 4);
    int*   deg    = (int*)carve((size_t)N * 4);
    int*   rowptr = (int*)carve((size_t)(N + 1) * 4);
    int*   cursor = (int*)carve((size_t)N * 4);
    int*   eidx   = (int*)carve((size_t)Etot * 4);
    float* colsum = (float*)carve(512 * 4);
    float* colsq  = (float*)carve(512 * 4);
    float* pool   = (float*)carve((size_t)G * 128 * 4);
    float* cnt    = (float*)carve((size_t)G * 4);
    (void)ws_size; (void)n_in;

    // ---- CSR by destination (shared across all 3 layers) ----
    zero_u32<<<(N + 255) / 256, 256, 0, stream>>>((unsigned int*)deg, N);
    degree_count<<<(Etot + 255) / 256, 256, 0, stream>>>(edst, deg, N, E);
    scan_rowptr<<<1, 1024, 0, stream>>>(deg, rowptr, N);
    copy_i32<<<(N + 255) / 256, 256, 0, stream>>>(rowptr, cursor, N);
    csr_fill<<<(Etot + 255) / 256, 256, 0, stream>>>(edst, cursor, eidx, N, E);

    auto run_layer = [&](const float* in, int K, int layer, int H) {
        const int OC = H * 128;
        const int grps = (N / 16) * (OC / 64);
        wmma_gemm_f32<<<(grps + 7) / 8, 256, 0, stream>>>(in, W[layer], bufA, N, K, OC);
        if (H == 4) {
            gat_scores<4><<<N, 128, 0, stream>>>(bufA, ASRC[layer], ADST[layer], ssrc, sdst, N);
            gat_aggregate<4><<<(N + 7) / 8, 256, 0, stream>>>(bufA, ssrc, sdst, rowptr, eidx,
                                                              esrc, bufB, N, E);
        } else {
            gat_scores<1><<<N, 32, 0, stream>>>(bufA, ASRC[layer], ADST[layer], ssrc, sdst, N);
            gat_aggregate<1><<<(N + 7) / 8, 256, 0, stream>>>(bufA, ssrc, sdst, rowptr, eidx,
                                                              esrc, bufB, N, E);
        }
        zero_u32<<<(OC + 255) / 256, 256, 0, stream>>>((unsigned int*)colsum, OC);
        zero_u32<<<(OC + 255) / 256, 256, 0, stream>>>((unsigned int*)colsq, OC);
        bn_stats<<<dim3(256, OC / 128), 128, 0, stream>>>(bufB, colsum, colsq, N, OC);
        size_t tot = (size_t)N * OC;
        bn_apply_elu<<<(unsigned)((tot + 255) / 256), 256, 0, stream>>>(bufB, colsum, colsq,
                                                                        GAM[layer], BET[layer], N, OC);
    };

    run_layer(x,    128, 0, 4);    // [N,128] -> [N,512]
    run_layer(bufB, 512, 1, 4);    // [N,512] -> [N,512]
    run_layer(bufB, 512, 2, 1);    // [N,512] -> [N,128]

    // ---- global mean pool ----
    zero_u32<<<(G * 128 + 255) / 256, 256, 0, stream>>>((unsigned int*)pool, G * 128);
    zero_u32<<<(G + 255) / 256, 256, 0, stream>>>((unsigned int*)cnt, G);
    pool_accum<<<N, 128, 0, stream>>>(bufB, batch, pool, cnt, N);
    pool_final<<<(G * 128 + 255) / 256, 256, 0, stream>>>(pool, cnt, (float*)d_out, G);
}